// S3FDNet_59133109732113
// MI455X (gfx1250) — compile-verified
//
#include <hip/hip_runtime.h>
#include <hip/hip_bf16.h>

// ---------------------------------------------------------------------------
// S3FD Detect post-process for MI455X (gfx1250, wave32).
//   K1: decode boxes + class-1 softmax score -> workspace (streaming, HBM-bound)
//   K2: per-batch (1 WG / batch, 1024 thr): exact top-5000 via LDS histogram
//       radix-select fed by a DOUBLE-BUFFERED TDM async pipeline, then a
//       750-step greedy NMS fully resident in LDS (~312KB of CDNA5's 320KB/WGP)
//       with wave32 shfl + ds_max_u64 argmax (2 barriers per serial round).
// ---------------------------------------------------------------------------

#define CONF_THRESH 0.05f
#define NMS_THRESH  0.3f
#define NMSK   5000      // NMS_TOP_K
#define TOPK   750       // TOP_K rows out per class
#define NBINS  4096      // 12-bit histogram (sign+exp8+mant3)
#define BINB_CAP 6144    // capacity for threshold-bin refinement list
#define CHUNK  16384     // score keys per TDM chunk (64KB)
#define NTHREADS 1024

typedef unsigned int uint4v __attribute__((ext_vector_type(4)));
typedef int          int4v  __attribute__((ext_vector_type(4)));
typedef int          int8v  __attribute__((ext_vector_type(8)));

#if defined(__HIP_DEVICE_COMPILE__) && defined(__has_builtin)
#if __has_builtin(__builtin_amdgcn_tensor_load_to_lds) && __has_builtin(__builtin_amdgcn_s_wait_tensorcnt)
#define HAVE_TDM 1
#endif
#endif

// Device-pass proof that the async-tensor path is the one being compiled.
#if defined(__HIP_DEVICE_COMPILE__) && !defined(HAVE_TDM)
#error "gfx1250 TDM builtins not available: tensor_load_to_lds path would not be emitted"
#endif

// Monotonic float->uint key (descending sort order == descending uint order).
__device__ __forceinline__ unsigned key_of(float f) {
  unsigned u = __float_as_uint(f);
  return (u & 0x80000000u) ? ~u : (u | 0x80000000u);
}
__device__ __forceinline__ float key_to_float(unsigned u) {
  return (u & 0x80000000u) ? __uint_as_float(u & 0x7FFFFFFFu)
                           : __uint_as_float(~u);
}

#if defined(HAVE_TDM)
// Issue one 1-D TDM DMA: n dwords from `gsrc` into LDS buffer `ldst`.
// `remaining` = dwords left in the tensor (tail OOB reads return zero).
__device__ __forceinline__ void tdm_issue_dwords(const unsigned* gsrc, int remaining,
                                                 int n, unsigned* ldst) {
  const unsigned long long ga = (unsigned long long)(const void*)gsrc;
  const unsigned lds_off = (unsigned)(unsigned long long)(void*)ldst;
  uint4v g0;
  g0.x = 1u;                                                  // count=1, user mode
  g0.y = lds_off;                                             // lds_addr
  g0.z = (unsigned)ga;                                        // global_addr[31:0]
  g0.w = (unsigned)((ga >> 32) & 0x01FFFFFFull) | (2u << 30); // addr[56:32] | type=2
  const unsigned dim0 = (unsigned)remaining;
  int8v g1 = {};
  g1[0] = (int)(2u << 16);                                    // data_size=4B, wg_mask=0
  g1[1] = (int)((dim0 & 0xFFFFu) << 16);                      // tensor_dim0[15:0]
  g1[2] = (int)((dim0 >> 16) | (1u << 16));                   // tensor_dim0[31:16] | tensor_dim1=1
  g1[3] = (int)((unsigned)n << 16);                           // tile_dim0
  g1[4] = (int)1u;                                            // tile_dim1=1, tile_dim2=0
  g1[5] = (int)dim0;                                          // tensor_dim0_stride[31:0]
  g1[6] = 0; g1[7] = 0;
  int4v z4 = {};
#if __clang_major__ >= 23
  int8v z8 = {};
  __builtin_amdgcn_tensor_load_to_lds(g0, g1, z4, z4, z8, 0);
#else
  __builtin_amdgcn_tensor_load_to_lds(g0, g1, z4, z4, 0);
#endif
}
#endif

// ---------------------------------------------------------------------------
// Kernel A: decode + score. One thread per (b,p). float4/float2 coalesced.
// ---------------------------------------------------------------------------
__global__ __launch_bounds__(256) void s3fd_decode_score(
    const float* __restrict__ loc, const float* __restrict__ conf,
    const float* __restrict__ prior, float4* __restrict__ boxes,
    unsigned* __restrict__ keys, int P) {
  const int p = blockIdx.x * 256 + threadIdx.x;
  const int b = blockIdx.y;
  if (p >= P) return;
  const size_t idx = (size_t)b * P + p;

  const float4 l  = ((const float4*)loc)[idx];
  const float2 c  = ((const float2*)conf)[idx];
  const float4 pr = ((const float4*)prior)[p];

  // decode (variances 0.1 / 0.2), corner form
  const float cx = pr.x + l.x * 0.1f * pr.z;
  const float cy = pr.y + l.y * 0.1f * pr.w;
  const float w  = pr.z * __expf(l.z * 0.2f);
  const float h  = pr.w * __expf(l.w * 0.2f);
  const float x1 = cx - 0.5f * w, y1 = cy - 0.5f * h;
  boxes[idx] = make_float4(x1, y1, x1 + w, y1 + h);

  // softmax class-1 prob == sigmoid(c1-c0); conf-threshold to -1 sentinel
  float s = 1.0f / (1.0f + __expf(c.x - c.y));
  s = (s > CONF_THRESH) ? s : -1.0f;
  keys[idx] = key_of(s);
}

// ---------------------------------------------------------------------------
// Kernel B: per-batch selection + NMS, one workgroup per batch.
// Dynamic LDS layout (16B aligned slabs), total 319536 B (< 327680 B/WGP):
// ---------------------------------------------------------------------------
enum : unsigned {
  OFF_CAND_B = 0,                      // float4 cand_b[5120]   81920
  OFF_CAND_S = 81920,                  // float  cand_s[5120]   20480
  OFF_CAND_A = 102400,                 // float  cand_a[5120]   20480
  OFF_HIST   = 122880,                 // uint   hist[4096]     16384
  OFF_BINB_K = 139264,                 // uint   binb_k[6144]   24576
  OFF_BINB_I = 163840,                 // uint   binb_i[6144]   24576
  OFF_KCH0   = 188416,                 // uint   kchunk0[16384] 65536
  OFF_KCH1   = 253952,                 // uint   kchunk1[16384] 65536
  OFF_CTL    = 319488,                 // uint   ctl[8]            32
  OFF_AMAX   = 319520,                 // u64    amax[2]           16
  SMEM_BYTES = 319536
};

__global__ __launch_bounds__(NTHREADS, 1) void s3fd_select_nms(
    const unsigned* __restrict__ keysBase, const float4* __restrict__ boxesBase,
    float* __restrict__ outBase, int P) {
  extern __shared__ __align__(16) unsigned char smem[];
  float4*   cand_b = (float4*)(smem + OFF_CAND_B);
  float*    cand_s = (float*)(smem + OFF_CAND_S);
  float*    cand_a = (float*)(smem + OFF_CAND_A);
  unsigned* hist   = (unsigned*)(smem + OFF_HIST);
  unsigned* binb_k = (unsigned*)(smem + OFF_BINB_K);
  unsigned* binb_i = (unsigned*)(smem + OFF_BINB_I);
  unsigned* kch0   = (unsigned*)(smem + OFF_KCH0);
  unsigned* kch1   = (unsigned*)(smem + OFF_KCH1);
  unsigned* ctl    = (unsigned*)(smem + OFF_CTL);  // 0:nCand 1:nBinB 2:thrBin 3:need
  unsigned long long* amax = (unsigned long long*)(smem + OFF_AMAX);

  const int tid = threadIdx.x;
  const int b   = blockIdx.x;
  const unsigned* keys  = keysBase  + (size_t)b * P;
  const float4*   boxes = boxesBase + (size_t)b * P;
  float*          out   = outBase   + (size_t)b * 2 * TOPK * 5;

  // Zero full output slab for this batch (class 0 stays zero; harness poisons
  // d_out so every call must rewrite it) and init LDS control/histogram.
  for (int i = tid; i < 2 * TOPK * 5; i += NTHREADS) out[i] = 0.0f;
  for (int i = tid; i < NBINS; i += NTHREADS) hist[i] = 0u;
  if (tid < 8) ctl[tid] = 0u;
  if (tid == 0) { amax[0] = 0ull; amax[1] = 0ull; }
  __syncthreads();

  // ---- Phase 1: 4096-bin histogram, double-buffered async TDM pipeline -----
  const int nchunks = (P + CHUNK - 1) / CHUNK;
#if defined(HAVE_TDM)
  if (tid < 32) tdm_issue_dwords(keys, P, (P < CHUNK ? P : CHUNK), kch0);
  for (int ci = 0; ci < nchunks; ++ci) {
    const int base = ci * CHUNK;
    const int n = (P - base < CHUNK) ? (P - base) : CHUNK;
    if (tid < 32) {
      if (ci + 1 < nchunks) {  // prefetch next chunk, then wait only for current
        const int nbase = base + CHUNK;
        const int nn = (P - nbase < CHUNK) ? (P - nbase) : CHUNK;
        tdm_issue_dwords(keys + nbase, P - nbase, nn, ((ci + 1) & 1) ? kch1 : kch0);
        __builtin_amdgcn_s_wait_tensorcnt(1);  // in-order: current chunk done
      } else {
        __builtin_amdgcn_s_wait_tensorcnt(0);
      }
    }
    __syncthreads();
    const unsigned* kc = (ci & 1) ? kch1 : kch0;
    for (int i = tid; i < n; i += NTHREADS)
      atomicAdd(&hist[kc[i] >> 20], 1u);
    __syncthreads();
  }
#else  // host parse pass only (device pass #errors out above if TDM missing)
  for (int ci = 0; ci < nchunks; ++ci) {
    const int base = ci * CHUNK;
    const int n = (P - base < CHUNK) ? (P - base) : CHUNK;
    for (int i = tid; i < n; i += NTHREADS) kch0[i] = keys[base + i];
    __syncthreads();
    for (int i = tid; i < n; i += NTHREADS) atomicAdd(&hist[kch0[i] >> 20], 1u);
    __syncthreads();
  }
#endif

  // ---- Phase 2: find threshold bin (rank NMSK from the top), exact need ----
  if (tid == 0) {
    unsigned cum = 0, thr = 0, need = NMSK;
    for (int bin = NBINS - 1; bin >= 0; --bin) {
      const unsigned c = hist[bin];
      if (cum + c >= NMSK) { thr = (unsigned)bin; need = NMSK - cum; break; }
      cum += c;
    }
    ctl[2] = thr; ctl[3] = need;
  }
  __syncthreads();
  const unsigned thr = ctl[2], need = ctl[3];

  // ---- Phase 3: compact strictly-above-threshold candidates into LDS, ------
  // ---- gather their boxes; collect threshold-bin members for refinement ----
  for (int i = tid; i < P; i += NTHREADS) {
    __builtin_prefetch(keys + i + NTHREADS * 8, 0, 0);  // global_prefetch_b8
    const unsigned u = keys[i];
    const unsigned bin = u >> 20;
    if (bin > thr) {
      const unsigned pos = atomicAdd(&ctl[0], 1u);  // guaranteed < NMSK
      const float4 bx = boxes[i];
      cand_s[pos] = key_to_float(u);
      cand_b[pos] = bx;
      cand_a[pos] = (bx.z - bx.x) * (bx.w - bx.y);
    } else if (bin == thr) {
      const unsigned pos = atomicAdd(&ctl[1], 1u);
      if (pos < BINB_CAP) { binb_k[pos] = u; binb_i[pos] = (unsigned)i; }
    }
  }
  __syncthreads();

  // ---- Phase 4: exact in-bin rank select (ties -> lower index, like top_k) --
  const int nb = (int)((ctl[1] < BINB_CAP) ? ctl[1] : BINB_CAP);
  for (int t = tid; t < nb; t += NTHREADS) {
    const unsigned u = binb_k[t], id = binb_i[t];
    unsigned rank = 0;
    for (int j = 0; j < nb; ++j) {
      const unsigned uj = binb_k[j];
      rank += (uj > u) || (uj == u && binb_i[j] < id);
    }
    if (rank < need) {
      const unsigned pos = atomicAdd(&ctl[0], 1u);
      if (pos < NMSK) {
        const float4 bx = boxes[id];
        cand_s[pos] = key_to_float(u);
        cand_b[pos] = bx;
        cand_a[pos] = (bx.z - bx.x) * (bx.w - bx.y);
      }
    }
  }
  __syncthreads();
  const int nc = (int)((ctl[0] < NMSK) ? ctl[0] : NMSK);
  for (int i = nc + tid; i < NMSK; i += NTHREADS) {
    cand_s[i] = -1.0f;
    cand_b[i] = make_float4(0.f, 0.f, 0.f, 0.f);
    cand_a[i] = 0.0f;
  }
  __syncthreads();

  // ---- Phase 5: serial greedy NMS, 750 rounds, all state in LDS ------------
  // Packed argmax key: (orderable score << 32) | ~idx  => max == (best score,
  // lowest index on ties), matching jnp.argmax over the candidate vector.
  const int lane = tid & 31;
  for (int k = 0; k < TOPK; ++k) {
    unsigned long long* slot = amax + (k & 1);
    unsigned long long m = 0ull;
    for (int c = tid; c < NMSK; c += NTHREADS) {
      const unsigned long long pk =
          ((unsigned long long)key_of(cand_s[c]) << 32) | (unsigned)(~c);
      if (pk > m) m = pk;
    }
    for (int off = 16; off > 0; off >>= 1) {   // wave32 shuffle reduce
      const unsigned long long o = __shfl_down(m, off, 32);
      if (o > m) m = o;
    }
    if (lane == 0) atomicMax(slot, m);          // ds_max_u64 across 32 waves
    __syncthreads();
    const unsigned long long pk = *slot;
    if (tid == 0) amax[(k + 1) & 1] = 0ull;     // reset next round's slot
    const int   sel = (int)(~(unsigned)pk);
    const float ss  = key_to_float((unsigned)(pk >> 32));
    if (!(ss > 0.0f)) break;  // remaining rows stay zero (out pre-zeroed)

    const float4 bb = cand_b[sel];
    const float  ab = cand_a[sel];
    for (int c = tid; c < NMSK; c += NTHREADS) {
      const float4 cb = cand_b[c];
      const float xx1 = fmaxf(cb.x, bb.x), yy1 = fmaxf(cb.y, bb.y);
      const float xx2 = fminf(cb.z, bb.z), yy2 = fminf(cb.w, bb.w);
      const float inter = fmaxf(xx2 - xx1, 0.0f) * fmaxf(yy2 - yy1, 0.0f);
      const float iou = inter / (cand_a[c] - inter + ab);
      if (iou > NMS_THRESH || c == sel) cand_s[c] = -1.0f;
    }
    if (tid == 0) {
      float* row = out + (size_t)(TOPK + k) * 5;  // class 1 slab
      row[0] = ss; row[1] = bb.x; row[2] = bb.y; row[3] = bb.z; row[4] = bb.w;
    }
    __syncthreads();
  }
}

// ---------------------------------------------------------------------------
extern "C" void kernel_launch(void* const* d_in, const int* in_sizes, int n_in,
                              void* d_out, int out_size, void* d_ws, size_t ws_size,
                              hipStream_t stream) {
  const float* loc   = (const float*)d_in[0];  // [B,P,4]
  const float* conf  = (const float*)d_in[1];  // [B,P,2]
  const float* prior = (const float*)d_in[2];  // [P,4]
  const int P = in_sizes[2] / 4;
  const int B = in_sizes[0] / (4 * P);

  // Workspace: boxes [B,P]float4 then keys [B,P]u32 (~21.9 MB for B=8,P=136500)
  float4*   boxes = (float4*)d_ws;
  unsigned* keys  = (unsigned*)((char*)d_ws + (size_t)B * P * sizeof(float4));

  dim3 gA((unsigned)((P + 255) / 256), (unsigned)B);
  s3fd_decode_score<<<gA, 256, 0, stream>>>(loc, conf, prior, boxes, keys, P);

  (void)hipFuncSetAttribute((const void*)s3fd_select_nms,
                            hipFuncAttributeMaxDynamicSharedMemorySize,
                            (int)SMEM_BYTES);
  s3fd_select_nms<<<dim3((unsigned)B), dim3(NTHREADS), SMEM_BYTES, stream>>>(
      keys, boxes, (float*)d_out, P);
  (void)n_in; (void)out_size; (void)ws_size;
}